// KeyedLayer_76862734729853
// MI455X (gfx1250) — compile-verified
//
#include <hip/hip_runtime.h>

// Problem constants (from the reference):
//   C=32, H=32, W=32, KS=3  ->  N = C*H*W + 1 = 32769, K = C*KS*KS = 288, B = 32
#define NN 32769
#define KK 288

// gfx1250 async global->LDS copy path (ASYNCcnt-tracked).
#if __has_builtin(__builtin_amdgcn_global_load_async_to_lds_b32) && \
    __has_builtin(__builtin_amdgcn_s_wait_asynccnt)
#define HAS_ASYNC_LDS 1
#else
#define HAS_ASYNC_LDS 0
#endif

// Exact parameter types per hipcc diagnostic: first arg is AS1 (global) int*,
// LDS side is AS3 (local) int*.
typedef __attribute__((address_space(1))) int* gas1_i32p;
typedef __attribute__((address_space(3))) int* as3_i32p;

// ---------------------------------------------------------------------------
// Kernel 1: transpose x (B x N, row-major) -> xt (N x B).
// Load phase uses GLOBAL_LOAD_ASYNC_TO_LDS_B32: per-lane global source and
// per-lane LDS destination, no VGPR round-trip; wave waits ASYNCcnt==0, then
// the workgroup barrier makes all waves' async writes visible before the
// transposed, fully-coalesced store phase.
// ---------------------------------------------------------------------------
__global__ __launch_bounds__(256) void xpose_kernel(const float* __restrict__ x,
                                                    float* __restrict__ xt) {
    __shared__ float tile[32][33];  // +1 pad: conflict-free transposed reads
    const int n0 = blockIdx.x * 32;
    const int tx = threadIdx.x;     // 0..31 : n offset on load, b on store
    const int ty = threadIdx.y;     // 0..7
    const int n_in = n0 + tx;

#if HAS_ASYNC_LDS
    if (n_in < NN) {
#pragma unroll
        for (int i = 0; i < 4; ++i) {
            const int b = ty + i * 8;
            __builtin_amdgcn_global_load_async_to_lds_b32(
                (gas1_i32p)(int*)(float*)const_cast<float*>(x + b * NN + n_in),
                (as3_i32p)(int*)(void*)(&tile[b][tx]),
                /*imm offset*/ 0, /*cpol*/ 0);
        }
    }
    __builtin_amdgcn_s_wait_asynccnt(0);  // this wave's async copies done
    __syncthreads();                      // all waves' copies visible
#else
    if (n_in < NN) {
#pragma unroll
        for (int i = 0; i < 4; ++i) {
            const int b = ty + i * 8;
            tile[b][tx] = x[b * NN + n_in];
        }
    }
    __syncthreads();
#endif

    // Store phase: tx now plays the role of b; 32 lanes write one contiguous
    // 128B line of xt per iteration. Tail-tile garbage entries are never
    // stored (guarded by n < NN), so no zero-fill is needed.
#pragma unroll
    for (int i = 0; i < 4; ++i) {
        const int r = ty + i * 8;       // n offset within tile
        const int n = n0 + r;
        if (n < NN) xt[n * 32 + tx] = tile[tx][r];
    }
}

// ---------------------------------------------------------------------------
// Kernel 2 (codegen verified in round 1): one wave per output row n.
// n = blockIdx.x is provably wave-uniform -> cols/vals rows stream through the
// scalar cache (s_load_bXXX, KMcnt), the gather is one 128B-line coalesced
// global_load_b32 ... scale_offset per (n,k), 16 independent loads in flight
// per wave, global_prefetch_b8 pre-pulls the HBM streams.
// ---------------------------------------------------------------------------
template <int MODE>
__global__ __launch_bounds__(32) void gatherdot_kernel(const int*   __restrict__ cols,
                                                       const float* __restrict__ vals,
                                                       const float* __restrict__ xsrc,
                                                       float*       __restrict__ out) {
    const int lane = threadIdx.x;       // lane == batch index b
    const int n    = blockIdx.x;        // uniform per wave
    if (n >= NN) return;

    const int*   __restrict__ crow = cols + n * KK;
    const float* __restrict__ vrow = vals + n * KK;

    float acc = 0.0f;
    for (int k0 = 0; k0 < KK; k0 += 16) {
        __builtin_prefetch(crow + k0 + 128, 0, 1);  // global_prefetch_b8
        __builtin_prefetch(vrow + k0 + 128, 0, 1);
#pragma unroll
        for (int kk = 0; kk < 16; ++kk) {
            const int   c = crow[k0 + kk];   // wave-uniform -> scalar load
            const float v = vrow[k0 + kk];   // wave-uniform -> scalar load
            float xv;
            if (MODE == 0) {
                xv = xsrc[(c << 5) | lane];  // xt (N x 32): one line per wave
            } else {
                xv = xsrc[lane * NN + c];    // direct fallback (no workspace)
            }
            acc = fmaf(v, xv, acc);
        }
    }
    out[lane * NN + n] = acc;  // y (B, N); 4B scatters merge in L2
}

// ---------------------------------------------------------------------------
extern "C" void kernel_launch(void* const* d_in, const int* in_sizes, int n_in,
                              void* d_out, int out_size, void* d_ws, size_t ws_size,
                              hipStream_t stream) {
    const float* x    = (const float*)d_in[0];   // x_affine (B, N) f32
    const float* vals = (const float*)d_in[1];   // vals     (N, K) f32
    const int*   cols = (const int*)  d_in[2];   // cols     (N, K) i32
    float*       out  = (float*)d_out;           // y        (B, N) f32

    const size_t xt_bytes = (size_t)NN * 32 * sizeof(float);  // ~4.2 MB

    if (ws_size >= xt_bytes) {
        float* xt = (float*)d_ws;
        dim3 tb(32, 8);
        xpose_kernel<<<(NN + 31) / 32, tb, 0, stream>>>(x, xt);
        gatherdot_kernel<0><<<NN, 32, 0, stream>>>(cols, vals, xt, out);
    } else {
        // Deterministic fallback (ws too small): gather straight from x.
        gatherdot_kernel<1><<<NN, 32, 0, stream>>>(cols, vals, x, out);
    }
}